// IRGCNModel_89558658056596
// MI455X (gfx1250) — compile-verified
//
#include <hip/hip_runtime.h>

typedef __attribute__((ext_vector_type(2))) float v2f;
typedef __attribute__((ext_vector_type(8))) float v8f;

#define HID 10
#define INV_MOD 0.1f

// Single-instruction relu: v_med3_f32(x, 0, +inf). (NaN -> 0; don't-care here.)
__device__ __forceinline__ float relu1(float x) {
    return __builtin_amdgcn_fmed3f(x, 0.0f, __builtin_inff());
}

// Native no-return f32 atomic add at device scope (L2 atomic units).
// Aggregation rows are shared across WGPs -> must be SCOPE_DEV.
__device__ __forceinline__ void atomic_add_f32_dev(float* p, float v) {
    asm volatile("global_atomic_add_f32 %0, %1, off scope:SCOPE_DEV"
                 :: "v"(p), "v"(v)
                 : "memory");
}

// ---------------------------------------------------------------------------
// Zero-fill (graph-capture-safe init of aggregation buffers)
// ---------------------------------------------------------------------------
__global__ void zero_kernel(float* __restrict__ p, int n) {
    int i = blockIdx.x * blockDim.x + threadIdx.x;
    if (i < n) p[i] = 0.0f;
}

// ---------------------------------------------------------------------------
// Edge / message-passing layer.
//   per edge e: m = relu(concat(h[dst], h[src]) @ W2); m = relu(m @ W1)/10
//   agg[dst] += m      (native f32 atomics into L2)
// One wave handles a tile of 16 edges using V_WMMA_F32_16X16X4_F32 chains.
// ---------------------------------------------------------------------------
template <int IN_DIM>
__global__ __launch_bounds__(128) void rgcn_edge_kernel(
    const float* __restrict__ Xin,   // [N, IN_DIM] (pre-activation for IN_DIM==HID)
    const float* __restrict__ W2,    // [2*IN_DIM, HID]
    const float* __restrict__ W1,    // [HID, HID]
    const int*   __restrict__ esrc,
    const int*   __restrict__ edst,
    float*       __restrict__ agg,   // [N, HID], pre-zeroed
    int nEdges)
{
    constexpr int KTOT   = 2 * IN_DIM;
    constexpr int NSTEP  = KTOT / 4;          // K-chain length for GEMM1
    constexpr bool RELU_IN = (IN_DIM == HID); // hidden layers gather relu(agg)
    constexpr int LSTR = 18;                  // LDS row stride (bank spread, 8B-aligned)

    __shared__ float ldsM[4][16 * LSTR];

    const int lane = threadIdx.x & 31;
    const int wv   = threadIdx.x >> 5;
    const int lr   = lane & 15;   // row (A/C) or col (B/C) index
    const int hi   = lane >> 4;   // K-half / M-half selector

    // Clamp+mask instead of divergent branches: EXEC stays full.
    const int   lw = (lr < HID) ? lr : 0;       // clamped weight column
    const float wm = (lr < HID) ? 1.0f : 0.0f;  // column validity mask

    // ---- B-fragments for W2 (KTOT x HID, N padded to 16), register resident.
    v2f w2f[NSTEP];
#pragma unroll
    for (int s = 0; s < NSTEP; ++s) {
        const int k = s * 4 + hi * 2;           // always < KTOT (KTOT % 4 == 0)
        w2f[s][0] = W2[k * HID + lw] * wm;
        w2f[s][1] = W2[(k + 1) * HID + lw] * wm;
    }
    // ---- B-fragments for W1 (10x10, K padded to 12, N padded to 16).
    v2f w1f[3];
#pragma unroll
    for (int s = 0; s < 3; ++s) {
        const int   k  = s * 4 + hi * 2;        // {0,2,4,6,8,10}; only 10 invalid
        const int   kc = (k < HID) ? k : 0;
        const float mm = ((k < HID) ? 1.0f : 0.0f) * wm;
        w1f[s][0] = W1[kc * HID + lw] * mm;
        w1f[s][1] = W1[(kc + 1) * HID + lw] * mm;
    }

    const int nTiles     = (nEdges + 15) / 16;
    const int waveId     = blockIdx.x * 4 + wv;
    const int waveStride = gridDim.x * 4;

    for (int tile = waveId; tile < nTiles; tile += waveStride) {
        const int ebase = tile * 16;

        // Prefetch next tile's edge indices (global_prefetch_b8, free).
        const int ntile = tile + waveStride;
        if (ntile < nTiles) {
            __builtin_prefetch(edst + ntile * 16, 0, 1);
            __builtin_prefetch(esrc + ntile * 16, 0, 1);
        }

        int eRow = ebase + lr;
        if (eRow >= nEdges) eRow = nEdges - 1;     // clamp; tail rows never scattered
        const int dRow = edst[eRow];
        const int sRow = esrc[eRow];

        // ---- GEMM1: A[16 x KTOT] (gathered rows) x W2 -> 16x16 (cols>=HID are 0)
        v8f acc = {};
#pragma unroll
        for (int s = 0; s < NSTEP; ++s) {
            const int k = s * 4 + hi * 2;
            // (k, k+1) never straddles IN_DIM (both even dims), so one select:
            const float* rowp = (k < IN_DIM) ? (Xin + (size_t)dRow * IN_DIM + k)
                                             : (Xin + (size_t)sRow * IN_DIM + (k - IN_DIM));
            v2f a = *(const v2f*)rowp;
            if (RELU_IN) { a[0] = relu1(a[0]); a[1] = relu1(a[1]); }
            acc = __builtin_amdgcn_wmma_f32_16x16x4_f32(
                false, a, false, w2f[s], (short)0, acc, false, false);
        }

        // ---- relu + D-layout -> A-layout transpose via per-wave LDS tile.
#pragma unroll
        for (int r = 0; r < 8; ++r) {
            const int m = r + hi * 8;
            ldsM[wv][m * LSTR + lr] = relu1(acc[r]);
        }
        asm volatile("s_wait_dscnt 0" ::: "memory");

        // ---- GEMM2: [16 x 12] x W1 -> 16x16
        v8f acc2 = {};
#pragma unroll
        for (int s = 0; s < 3; ++s) {
            const int k = s * 4 + hi * 2;
            v2f a = *(const v2f*)(&ldsM[wv][lr * LSTR + k]);
            acc2 = __builtin_amdgcn_wmma_f32_16x16x4_f32(
                false, a, false, w1f[s], (short)0, acc2, false, false);
        }

        // ---- relu, /MOD, atomic scatter-add to agg[dst].
        if (lr < HID) {
#pragma unroll
            for (int r = 0; r < 8; ++r) {
                const int m = r + hi * 8;
                const int e = ebase + m;
                if (e < nEdges) {
                    const int d = edst[e];
                    const float v = relu1(acc2[r]) * INV_MOD;
                    atomic_add_f32_dev(agg + (size_t)d * HID + lr, v);
                }
            }
        }
    }
}

// ---------------------------------------------------------------------------
// Final node MLP: out = relu(relu(relu(H) @ Wf1) @ Wf2)
// One wave handles 16 nodes; 12 + 64 WMMAs per tile.
// ---------------------------------------------------------------------------
__global__ __launch_bounds__(128) void rgcn_final_kernel(
    const float* __restrict__ H,    // [N, HID], pre-activation sums
    const float* __restrict__ Wf1,  // [HID, 64]
    const float* __restrict__ Wf2,  // [64, 64]
    float*       __restrict__ out,  // [N, 64]
    int nNodes)
{
    constexpr int LSTR = 66;  // 16x64 tile, padded row stride (bank spread)
    __shared__ float ldsT[4][16 * LSTR];

    const int lane = threadIdx.x & 31;
    const int wv   = threadIdx.x >> 5;
    const int lr   = lane & 15;
    const int hi   = lane >> 4;

    const int nTiles = (nNodes + 15) / 16;
    for (int tile = blockIdx.x * 4 + wv; tile < nTiles; tile += gridDim.x * 4) {
        const int nbase = tile * 16;
        int node = nbase + lr;
        if (node >= nNodes) node = nNodes - 1;

        // ---- A-fragments of relu(H), K padded 10 -> 12 (clamp+mask, no branches).
        v2f aH[3];
#pragma unroll
        for (int s = 0; s < 3; ++s) {
            const int   k  = s * 4 + hi * 2;     // {0,2,4,6,8,10}; only 10 invalid
            const int   kc = (k < HID) ? k : 0;
            const float km = (k < HID) ? 1.0f : 0.0f;
            v2f t = *(const v2f*)(H + (size_t)node * HID + kc);
            aH[s][0] = relu1(t[0]) * km;
            aH[s][1] = relu1(t[1]) * km;
        }

        // ---- GEMM1: [16x12] x Wf1 -> 16x64, relu, stage into LDS.
#pragma unroll
        for (int ct = 0; ct < 4; ++ct) {
            v8f acc = {};
#pragma unroll
            for (int s = 0; s < 3; ++s) {
                const int   k  = s * 4 + hi * 2;
                const int   kc = (k < HID) ? k : 0;
                const float km = (k < HID) ? 1.0f : 0.0f;
                v2f b;
                b[0] = Wf1[kc * 64 + ct * 16 + lr] * km;
                b[1] = Wf1[(kc + 1) * 64 + ct * 16 + lr] * km;
                acc = __builtin_amdgcn_wmma_f32_16x16x4_f32(
                    false, aH[s], false, b, (short)0, acc, false, false);
            }
#pragma unroll
            for (int r = 0; r < 8; ++r) {
                const int m = r + hi * 8;
                ldsT[wv][m * LSTR + ct * 16 + lr] = relu1(acc[r]);
            }
        }
        asm volatile("s_wait_dscnt 0" ::: "memory");

        // ---- GEMM2: [16x64] x Wf2 -> 16x64, relu, store.
#pragma unroll
        for (int ct = 0; ct < 4; ++ct) {
            v8f acc = {};
#pragma unroll
            for (int s = 0; s < 16; ++s) {
                const int k = s * 4 + hi * 2;
                v2f a = *(const v2f*)(&ldsT[wv][lr * LSTR + k]);
                v2f b;
                b[0] = Wf2[k * 64 + ct * 16 + lr];
                b[1] = Wf2[(k + 1) * 64 + ct * 16 + lr];
                acc = __builtin_amdgcn_wmma_f32_16x16x4_f32(
                    false, a, false, b, (short)0, acc, false, false);
            }
#pragma unroll
            for (int r = 0; r < 8; ++r) {
                const int m  = r + hi * 8;
                const int nd = nbase + m;
                if (nd < nNodes)
                    out[(size_t)nd * 64 + ct * 16 + lr] = relu1(acc[r]);
            }
        }
    }
}

// ---------------------------------------------------------------------------
// Launch
// ---------------------------------------------------------------------------
extern "C" void kernel_launch(void* const* d_in, const int* in_sizes, int n_in,
                              void* d_out, int out_size, void* d_ws, size_t ws_size,
                              hipStream_t stream) {
    const float* X    = (const float*)d_in[0];
    const float* W2_0 = (const float*)d_in[1];
    const float* W1_0 = (const float*)d_in[2];
    const float* W2_1 = (const float*)d_in[3];
    const float* W1_1 = (const float*)d_in[4];
    const float* W2_2 = (const float*)d_in[5];
    const float* W1_2 = (const float*)d_in[6];
    const float* Wf1  = (const float*)d_in[7];
    const float* Wf2  = (const float*)d_in[8];
    const int*   esrc = (const int*)d_in[9];
    const int*   edst = (const int*)d_in[10];
    float* out = (float*)d_out;

    const int nNodes = in_sizes[0] / 64;  // FEAT = 64
    const int nEdges = in_sizes[9];

    float* bufA = (float*)d_ws;                      // [N, HID]
    float* bufB = bufA + (size_t)nNodes * HID;       // [N, HID]
    const int aggN = nNodes * HID;

    const dim3 zb(256), zg((aggN + 255) / 256);
    const int EDGE_BLOCKS = 4096;   // x4 waves, grid-stride over 16-edge tiles
    const int NODE_BLOCKS = 1024;   // x4 waves, grid-stride over 16-node tiles

    // Layer 0: X[64] -> bufA
    zero_kernel<<<zg, zb, 0, stream>>>(bufA, aggN);
    rgcn_edge_kernel<64><<<EDGE_BLOCKS, 128, 0, stream>>>(X, W2_0, W1_0, esrc, edst, bufA, nEdges);
    // Layer 1: relu(bufA)[10] -> bufB
    zero_kernel<<<zg, zb, 0, stream>>>(bufB, aggN);
    rgcn_edge_kernel<HID><<<EDGE_BLOCKS, 128, 0, stream>>>(bufA, W2_1, W1_1, esrc, edst, bufB, nEdges);
    // Layer 2: relu(bufB)[10] -> bufA
    zero_kernel<<<zg, zb, 0, stream>>>(bufA, aggN);
    rgcn_edge_kernel<HID><<<EDGE_BLOCKS, 128, 0, stream>>>(bufB, W2_2, W1_2, esrc, edst, bufA, nEdges);
    // Final MLP: relu(bufA) @ Wf1 @ Wf2 -> out
    rgcn_final_kernel<<<NODE_BLOCKS, 128, 0, stream>>>(bufA, Wf1, Wf2, out, nNodes);
}